// Detector_33071248179852
// MI455X (gfx1250) — compile-verified
//
#include <hip/hip_runtime.h>
#include <cstdint>
#include <cstddef>

// ---------------- problem constants (match reference) ----------------
#define N_CLS   365
#define N_SEL   4256          // 1000+1000+1000+1000+256
#define SORT_N  8192
#define SCORE_TH 0.05f
#define IOU_TH   0.6f

typedef unsigned v4u_ __attribute__((ext_vector_type(4)));
typedef int      v4i_ __attribute__((ext_vector_type(4)));
typedef int      v8i_ __attribute__((ext_vector_type(8)));

#if defined(__has_builtin)
#if __has_builtin(__builtin_amdgcn_tensor_load_to_lds)
#define HAVE_TDM 1
#else
#define HAVE_TDM 0
#endif
#else
#define HAVE_TDM 0
#endif

// =====================================================================
// Kernel 1: per-row max/argmax of logits.  The 8x365 f32 tile (row
// stride 365) is moved global->LDS by the Tensor Data Mover (one
// tensor_load_to_lds per tile, TENSORcnt-tracked), double-buffered so
// tile k+1 streams while tile k is reduced.  sigmoid applied once/row.
// =====================================================================
#define ROWS_TILE 8

#if HAVE_TDM
__device__ __forceinline__ void tdm_issue_tile(const float* base, long long R,
                                               int rows, const void* lds_dst)
{
    unsigned long long ga = (unsigned long long)(uintptr_t)base +
                            (unsigned long long)R * (unsigned long long)N_CLS * 4ull;
    unsigned lds = (unsigned)(uintptr_t)lds_dst;

    // D# group 0 (128b): count=1 | lds_addr | global_addr[56:0] | type=2
    v4u_ g0;
    g0[0] = 1u;                                            // count=1, user descriptor
    g0[1] = lds;                                           // lds_addr (bytes)
    g0[2] = (unsigned)(ga & 0xFFFFFFFFull);                // global_addr[31:0]
    g0[3] = (unsigned)((ga >> 32) & 0x1FFFFFFull) | 0x80000000u; // ga[56:32] | type=2<<30

    // D# group 1 (256b): 2D tensor, 4-byte elements, no pad/iterate/multicast
    v8i_ g1;
    g1[0] = 0x00020000;                                    // data_size=2 (4B)
    g1[1] = (int)(((unsigned)N_CLS & 0xFFFFu) << 16);      // tensor_dim0 lo16
    g1[2] = (int)((N_CLS >> 16) | (((unsigned)rows & 0xFFFFu) << 16)); // dim0 hi | tensor_dim1 lo
    g1[3] = (int)((rows >> 16) | ((unsigned)N_CLS << 16)); // dim1 hi | tile_dim0
    g1[4] = rows;                                          // tile_dim1 (tile_dim2=0)
    g1[5] = N_CLS;                                         // tensor_dim0_stride lo32
    g1[6] = 0;                                             // stride hi | dim1_stride lo
    g1[7] = 0;

    v4i_ z4 = {0, 0, 0, 0};
    v8i_ z8 = {0, 0, 0, 0, 0, 0, 0, 0};
#if __clang_major__ >= 23
    __builtin_amdgcn_tensor_load_to_lds(g0, g1, z4, z4, z8, 0);
#else
    __builtin_amdgcn_tensor_load_to_lds(g0, g1, z4, z4, 0);
#endif
}
#else
// Fallback: per-lane CDNA5 async global->LDS copies (ASYNCcnt-tracked),
// issued by wave 0 only, packed layout identical to the TDM path.
__device__ __forceinline__ void async_issue_tile(const float* base, long long R,
                                                 int rows, const void* lds_dst,
                                                 int lane)
{
    const int nf = rows * N_CLS;
    const unsigned l0 = (unsigned)(uintptr_t)lds_dst;
    const float* g0 = base + R * (long long)N_CLS;
    for (int c = lane; c < nf; c += 32) {
        unsigned long long ga = (unsigned long long)(uintptr_t)(g0 + c);
        unsigned l = l0 + 4u * (unsigned)c;
        asm volatile("global_load_async_to_lds_b32 %0, %1, off"
                     :: "v"(l), "v"(ga) : "memory");
    }
}
#endif

__global__ __launch_bounds__(256)
void det_rowstat_kernel(const float* __restrict__ cls_logits,
                        const float* __restrict__ ctr,
                        float* __restrict__ p_all,
                        int*   __restrict__ cls_all,
                        int n_rows)
{
    __shared__ float buf[2][ROWS_TILE * N_CLS];
    const int tid  = threadIdx.x;
    const int lane = tid & 31;
    const int wid  = tid >> 5;

    long long R = (long long)blockIdx.x * ROWS_TILE;
    if (R >= n_rows) return;
    const long long step = (long long)gridDim.x * ROWS_TILE;

#if HAVE_TDM
    if (wid == 0) {
        long long rem = (long long)n_rows - R;
        tdm_issue_tile(cls_logits, R, (int)(rem < ROWS_TILE ? rem : ROWS_TILE), &buf[0][0]);
    }
#endif

    int cur = 0;
    for (; R < n_rows; R += step, cur ^= 1) {
        long long rem = (long long)n_rows - R;
        const int rows = (int)(rem < ROWS_TILE ? rem : ROWS_TILE);
#if HAVE_TDM
        const long long Rn = R + step;
        if (wid == 0) {
            if (Rn < n_rows) {
                long long remn = (long long)n_rows - Rn;
                tdm_issue_tile(cls_logits, Rn,
                               (int)(remn < ROWS_TILE ? remn : ROWS_TILE),
                               &buf[cur ^ 1][0]);
                __builtin_amdgcn_s_wait_tensorcnt(1);   // tile for THIS iter done (in-order)
            } else {
                __builtin_amdgcn_s_wait_tensorcnt(0);
            }
        }
#else
        if (wid == 0) {
            async_issue_tile(cls_logits, R, rows, &buf[cur][0], lane);
            asm volatile("s_wait_asynccnt 0" ::: "memory");
        }
#endif
        __syncthreads();                                // tile visible to all waves

        if (wid < rows) {
            const long long r = R + wid;
            const float* rowbuf = &buf[cur][wid * N_CLS];
            float m = -3.0e38f; int am = 0;
            for (int c2 = lane; c2 < N_CLS; c2 += 32) {
                float v = rowbuf[c2];
                if (v > m) { m = v; am = c2; }          // strict: first index wins in-lane
            }
            for (int s = 16; s; s >>= 1) {              // wave32 reduce, min-index ties
                float om = __shfl_xor(m, s, 32);
                int   oa = __shfl_xor(am, s, 32);
                if (om > m || (om == m && oa < am)) { m = om; am = oa; }
            }
            if (lane == 0) {
                float pr = 1.0f / (1.0f + expf(-m));    // sigmoid(max) == max(sigmoid)
                p_all[r]   = pr * ctr[r];
                cls_all[r] = am + 1;
            }
        }
        __syncthreads();                                // protect buf before next issue
    }
}

// =====================================================================
// Kernel 2: exact per-level top-k via 4-pass radix select (one block/level)
// scores >= 0 -> IEEE bits are order-preserving as uint32.
// =====================================================================
__global__ __launch_bounds__(1024)
void det_topk_kernel(const float* __restrict__ p_all, int* __restrict__ sel)
{
    const int starts[5] = {0, 65536, 81920, 86016, 87040};
    const int sizes[5]  = {65536, 16384, 4096, 1024, 256};
    const int ks[5]     = {1000, 1000, 1000, 1000, 256};
    const int outs[5]   = {0, 1000, 2000, 3000, 4000};

    const int lvl  = blockIdx.x;
    const int base = starts[lvl], num = sizes[lvl];
    const int k    = ks[lvl],     ob  = outs[lvl];

    __shared__ unsigned hist[256];
    __shared__ unsigned sh_prefix, sh_rem, cntG, cntE;
    const int tid = threadIdx.x, nt = blockDim.x;

    if (tid == 0) { sh_prefix = 0u; sh_rem = (unsigned)k; }
    __syncthreads();

    for (int pass = 0; pass < 4; ++pass) {
        const int shift = 24 - 8 * pass;
        const unsigned maskHi = (pass == 0) ? 0u : (0xFFFFFFFFu << (shift + 8));
        for (int b = tid; b < 256; b += nt) hist[b] = 0u;
        __syncthreads();
        const unsigned prefix = sh_prefix;
        for (int i = tid; i < num; i += nt) {
            unsigned key = __float_as_uint(p_all[base + i]);
            if (((key ^ prefix) & maskHi) == 0u)
                atomicAdd(&hist[(key >> shift) & 255u], 1u);
        }
        __syncthreads();
        if (tid == 0) {
            unsigned rem = sh_rem, cum = 0u; int b = 0;
            for (b = 255; b >= 0; --b) {
                cum += hist[b];
                if (cum >= rem) { sh_rem = rem - (cum - hist[b]); break; }
            }
            if (b < 0) b = 0;
            sh_prefix = prefix | ((unsigned)b << shift);
        }
        __syncthreads();
    }

    const unsigned T      = sh_prefix;
    const unsigned needEq = sh_rem;                 // equals to take
    const unsigned kG     = (unsigned)k - needEq;   // strictly-greater count
    if (tid == 0) { cntG = 0u; cntE = 0u; }
    __syncthreads();
    for (int i = tid; i < num; i += nt) {
        unsigned key = __float_as_uint(p_all[base + i]);
        if (key > T) {
            unsigned pos = atomicAdd(&cntG, 1u);
            sel[ob + pos] = base + i;
        } else if (key == T) {
            unsigned e = atomicAdd(&cntE, 1u);
            if (e < needEq) sel[ob + kG + e] = base + i;
        }
    }
}

// =====================================================================
// Kernel 3: gather selected entries + box crop/unpad/unscale
// =====================================================================
__global__ __launch_bounds__(256)
void det_gather_kernel(const int* __restrict__ sel,
                       const float* __restrict__ p_all,
                       const int*   __restrict__ cls_all,
                       const float* __restrict__ reg,
                       const float* __restrict__ loc,
                       float* __restrict__ sc, int* __restrict__ cl,
                       float* __restrict__ bx)
{
    int i = blockIdx.x * blockDim.x + threadIdx.x;
    if (i >= N_SEL) return;
    int idx = sel[i];
    sc[i] = p_all[idx];
    cl[i] = cls_all[idx];
    float px = loc[0], py = loc[1], cw = loc[2], ch = loc[3], s = loc[4];
    float r0 = reg[4 * idx + 0], r1 = reg[4 * idx + 1];
    float r2 = reg[4 * idx + 2], r3 = reg[4 * idx + 3];
    bx[4 * i + 0] = fmaxf(r0 - px, 0.0f) / s;
    bx[4 * i + 1] = fmaxf(r1 - py, 0.0f) / s;
    bx[4 * i + 2] = (fminf(r2, cw) - px) / s;
    bx[4 * i + 3] = (fminf(r3, ch) - py) / s;
}

// =====================================================================
// Kernel 4: single-block bitonic sort by score desc (invalid -> tail),
// scores staged global->LDS with CDNA5 async copies (ASYNCcnt), plus
// max over all box coords (for the class-offset trick).
// =====================================================================
__global__ __launch_bounds__(1024)
void det_sort_kernel(const float* __restrict__ sc,
                     const float* __restrict__ bx,
                     int* __restrict__ order,
                     float* __restrict__ boxmax)
{
    __shared__ float skey[SORT_N];
    __shared__ int   sidx[SORT_N];
    __shared__ float lsc[N_SEL];
    const int tid = threadIdx.x, nt = blockDim.x;

    // async-stage the contiguous score array into LDS (16B per lane per op)
    const unsigned lbase = (unsigned)(uintptr_t)(&lsc[0]);
    for (int c = tid; c < N_SEL / 4; c += nt) {
        unsigned long long ga = (unsigned long long)(uintptr_t)(sc + 4 * c);
        unsigned l = lbase + 16u * (unsigned)c;
        asm volatile("global_load_async_to_lds_b128 %0, %1, off"
                     :: "v"(l), "v"(ga) : "memory");
    }
    asm volatile("s_wait_asynccnt 0" ::: "memory");
    __syncthreads();

    for (int i = tid; i < SORT_N; i += nt) {
        float key;
        if (i < N_SEL) { float p = lsc[i]; key = (p > SCORE_TH) ? p : -1.0e30f; }
        else key = -2.0e30f;
        skey[i] = key; sidx[i] = i;
    }
    __syncthreads();

    for (int ksz = 2; ksz <= SORT_N; ksz <<= 1) {
        for (int j = ksz >> 1; j > 0; j >>= 1) {
            for (int t = tid; t < SORT_N / 2; t += nt) {
                int i   = 2 * t - (t & (j - 1));
                int ixj = i + j;
                bool desc = ((i & ksz) == 0);
                float a = skey[i], b = skey[ixj];
                bool sw = desc ? (a < b) : (a > b);
                if (sw) {
                    skey[i] = b; skey[ixj] = a;
                    int ta = sidx[i]; sidx[i] = sidx[ixj]; sidx[ixj] = ta;
                }
            }
            __syncthreads();
        }
    }

    for (int i = tid; i < N_SEL; i += nt) order[i] = sidx[i];

    float m = -3.0e38f;
    for (int i = tid; i < 4 * N_SEL; i += nt) m = fmaxf(m, bx[i]);
    skey[tid] = m;
    __syncthreads();
    for (int s = nt >> 1; s; s >>= 1) {
        if (tid < s) skey[tid] = fmaxf(skey[tid], skey[tid + s]);
        __syncthreads();
    }
    if (tid == 0) boxmax[0] = skey[0];
}

// =====================================================================
// Kernel 5: sequential greedy per-class NMS (class offset) + outputs
// =====================================================================
__global__ __launch_bounds__(1024)
void det_nms_kernel(const int* __restrict__ order,
                    const float* __restrict__ sc,
                    const int*   __restrict__ cl,
                    const float* __restrict__ bx,
                    const float* __restrict__ boxmax,
                    float* __restrict__ out)
{
    const int M = N_SEL;
    __shared__ float lx1[N_SEL], ly1[N_SEL], lx2[N_SEL], ly2[N_SEL];
    __shared__ unsigned char lkeep[N_SEL];
    const int tid = threadIdx.x, nt = blockDim.x;

    const float offsc = boxmax[0] + 1.0f;
    for (int i = tid; i < M; i += nt) {
        int j = order[i];
        float o = offsc * (float)cl[j];
        lx1[i] = bx[4 * j + 0] + o;
        ly1[i] = bx[4 * j + 1] + o;
        lx2[i] = bx[4 * j + 2] + o;
        ly2[i] = bx[4 * j + 3] + o;
        lkeep[i] = (sc[j] > SCORE_TH) ? 1 : 0;
    }
    __syncthreads();

    for (int i = 0; i < M; ++i) {
        if (lkeep[i]) {                               // uniform branch (LDS broadcast)
            float a = lx1[i], b = ly1[i], c = lx2[i], d = ly2[i];
            float ai = fmaxf(c - a, 0.0f) * fmaxf(d - b, 0.0f);
            for (int j2 = i + 1 + tid; j2 < M; j2 += nt) {
                if (!lkeep[j2]) continue;
                float xx1 = fmaxf(a, lx1[j2]);
                float yy1 = fmaxf(b, ly1[j2]);
                float xx2 = fminf(c, lx2[j2]);
                float yy2 = fminf(d, ly2[j2]);
                float aj  = fmaxf(lx2[j2] - lx1[j2], 0.0f) *
                            fmaxf(ly2[j2] - ly1[j2], 0.0f);
                float inter = fmaxf(xx2 - xx1, 0.0f) * fmaxf(yy2 - yy1, 0.0f);
                float iou = inter / (ai + aj - inter + 1e-9f);
                if (iou > IOU_TH) lkeep[j2] = 0;
            }
        }
        __syncthreads();                              // publish suppressions for i+1
    }

    for (int i = tid; i < M; i += nt) {
        int j = order[i];
        float kf = lkeep[i] ? 1.0f : 0.0f;
        out[i]                 = kf * (float)cl[j];
        out[M + i]             = kf * sc[j];
        out[2 * M + 4 * i + 0] = kf * bx[4 * j + 0];
        out[2 * M + 4 * i + 1] = kf * bx[4 * j + 1];
        out[2 * M + 4 * i + 2] = kf * bx[4 * j + 2];
        out[2 * M + 4 * i + 3] = kf * bx[4 * j + 3];
        out[6 * M + i]         = kf;
    }
}

// =====================================================================
// host launcher
// =====================================================================
extern "C" void kernel_launch(void* const* d_in, const int* in_sizes, int n_in,
                              void* d_out, int out_size, void* d_ws, size_t ws_size,
                              hipStream_t stream)
{
    (void)n_in; (void)out_size; (void)ws_size;
    const float* pred_cls  = (const float*)d_in[0];
    const float* pred_reg  = (const float*)d_in[1];
    const float* pred_ctr  = (const float*)d_in[2];
    const float* locations = (const float*)d_in[3];
    const int n_rows = in_sizes[2];                 // 87296

    char* ws = (char*)d_ws;
    size_t off = 0;
    auto bump = [&](size_t bytes) -> void* {
        void* p = ws + off;
        off += (bytes + 255) & ~(size_t)255;
        return p;
    };
    float* p_all   = (float*)bump((size_t)n_rows * sizeof(float));
    int*   cls_all = (int*)  bump((size_t)n_rows * sizeof(int));
    int*   sel     = (int*)  bump(N_SEL * sizeof(int));
    float* sc      = (float*)bump(N_SEL * sizeof(float));
    int*   cl      = (int*)  bump(N_SEL * sizeof(int));
    float* bx      = (float*)bump(4 * N_SEL * sizeof(float));
    int*   order   = (int*)  bump(N_SEL * sizeof(int));
    float* boxmax  = (float*)bump(sizeof(float));

    const int tiles = (n_rows + ROWS_TILE - 1) / ROWS_TILE;
    const int grid  = tiles < 4096 ? tiles : 4096;   // persistent blocks -> TDM prefetch overlaps
    det_rowstat_kernel<<<grid, 256, 0, stream>>>(pred_cls, pred_ctr,
                                                 p_all, cls_all, n_rows);
    det_topk_kernel<<<5, 1024, 0, stream>>>(p_all, sel);
    det_gather_kernel<<<(N_SEL + 255) / 256, 256, 0, stream>>>(
        sel, p_all, cls_all, pred_reg, locations, sc, cl, bx);
    det_sort_kernel<<<1, 1024, 0, stream>>>(sc, bx, order, boxmax);
    det_nms_kernel<<<1, 1024, 0, stream>>>(order, sc, cl, bx, boxmax, (float*)d_out);
}